// TripletLoss_19275813224966
// MI455X (gfx1250) — compile-verified
//
#include <hip/hip_runtime.h>

#define NROWS 8192
#define DDIM 64
#define NCLS 20
#define NPAD 32            // classes padded to 2 WMMA N-tiles
#define LSTRIDE 68         // LDS row stride (floats); 68 % 64 == 4 -> no bank conflicts
#define MARGIN 1.0f

typedef float v2f __attribute__((ext_vector_type(2)));
typedef float v8f __attribute__((ext_vector_type(8)));

__global__ void tl_init_ws(unsigned* __restrict__ P, unsigned* __restrict__ M,
                           int* __restrict__ cnt) {
    int c = threadIdx.x;            // 32 threads
    P[c]   = 0u;                    // bits of 0.0f (max identity; mask zeros included anyway)
    M[c]   = 0x7f800000u;           // bits of +inf (min identity)
    cnt[c] = 0;
}

__global__ void tl_count(const int* __restrict__ labels, int* __restrict__ cnt) {
    int i = blockIdx.x * blockDim.x + threadIdx.x;
    if (i < NROWS) atomicAdd(&cnt[labels[i]], 1);
}

// 64 blocks x 256 threads; each wave handles a 16-row tile vs all 32 padded classes.
__global__ void __launch_bounds__(256)
tl_dist_reduce(const float* __restrict__ emb, const int* __restrict__ labels,
               const float* __restrict__ labemb,
               unsigned* __restrict__ P, unsigned* __restrict__ M) {
    __shared__ float    sL[NPAD * LSTRIDE];   // padded label embeddings [class][k]
    __shared__ float    sql[NPAD];            // |L_c|^2
    __shared__ float    sqe[128];             // |e_i|^2 for this block's 128 rows
    __shared__ unsigned sP[NPAD];
    __shared__ unsigned sM[NPAD];

    const int tid = threadIdx.x;

    // Stage label embeddings into LDS, zero-padding classes 20..31.
    for (int idx = tid; idx < NPAD * DDIM; idx += 256) {
        int c = idx >> 6, k = idx & 63;
        sL[c * LSTRIDE + k] = (c < NCLS) ? labemb[c * DDIM + k] : 0.0f;
    }
    if (tid < NPAD) { sP[tid] = 0u; sM[tid] = 0x7f800000u; }
    __syncthreads();

    if (tid < NPAD) {
        float s = 0.0f;
        for (int k = 0; k < DDIM; ++k) { float v = sL[tid * LSTRIDE + k]; s += v * v; }
        sql[tid] = s;
    }

    const int wave    = tid >> 5;
    const int lane    = tid & 31;
    const int r       = lane & 15;          // row within tile == column (class) index
    const int half    = lane >> 4;          // selects K-pair (A) / K-pair (B)
    const int rowBase = blockIdx.x * 128 + wave * 16;

    // Per-row squared norms (lanes 0..15 of each wave, one row each).
    if (half == 0) {
        const float* row = emb + (size_t)(rowBase + r) * DDIM;
        float s = 0.0f;
        for (int k = 0; k < DDIM; ++k) { float v = row[k]; s += v * v; }
        sqe[wave * 16 + r] = s;
    }
    __syncthreads();

    // D = 64 -> 16 K-steps of WMMA f32 16x16x4; two N-tiles (classes 0-15, 16-31).
    v8f acc0 = {}; v8f acc1 = {};
    const float* arow = emb + (size_t)(rowBase + r) * DDIM + half * 2;
    const float* b0p  = &sL[r * LSTRIDE + half * 2];
    const float* b1p  = &sL[(16 + r) * LSTRIDE + half * 2];
    #pragma unroll
    for (int t = 0; t < 16; ++t) {
        const int k = 4 * t;
        v2f a  = *(const v2f*)(arow + k);                 // A[row][k..k+1] (K-pair per half)
        v2f b0 = { b0p[k], b0p[k + 1] };                  // B[k][n] = L[n][k]
        v2f b1 = { b1p[k], b1p[k + 1] };
        acc0 = __builtin_amdgcn_wmma_f32_16x16x4_f32(false, a, false, b0,
                                                     (short)0, acc0, false, false);
        acc1 = __builtin_amdgcn_wmma_f32_16x16x4_f32(false, a, false, b1,
                                                     (short)0, acc1, false, false);
    }

    // C/D layout: VGPR v, lane -> row m = v + 8*half, col n = lane&15.
    const int   n     = r;
    const float sqln0 = sql[n];
    const float sqln1 = sql[16 + n];
    unsigned lp0 = 0u, lm0 = 0x7f800000u, lp1 = 0u, lm1 = 0x7f800000u;
    #pragma unroll
    for (int v = 0; v < 8; ++v) {
        const int   m   = v + 8 * half;
        const int   lab = labels[rowBase + m];
        const float se  = sqe[wave * 16 + m];

        float d0 = fmaxf(se + sqln0 - 2.0f * acc0[v], 0.0f);
        unsigned p0 = __float_as_uint((lab == n)      ? d0 : 0.0f);
        unsigned q0 = __float_as_uint((lab == n)      ? 0.0f : d0);
        lp0 = lp0 > p0 ? lp0 : p0;  lm0 = lm0 < q0 ? lm0 : q0;

        float d1 = fmaxf(se + sqln1 - 2.0f * acc1[v], 0.0f);
        unsigned p1 = __float_as_uint((lab == 16 + n) ? d1 : 0.0f);
        unsigned q1 = __float_as_uint((lab == 16 + n) ? 0.0f : d1);
        lp1 = lp1 > p1 ? lp1 : p1;  lm1 = lm1 < q1 ? lm1 : q1;
    }
    // Nonnegative floats: IEEE order == unsigned-bit order -> integer atomics, deterministic.
    atomicMax(&sP[n], lp0);      atomicMin(&sM[n], lm0);
    atomicMax(&sP[16 + n], lp1); atomicMin(&sM[16 + n], lm1);
    __syncthreads();

    if (tid < NCLS) { atomicMax(&P[tid], sP[tid]); atomicMin(&M[tid], sM[tid]); }
}

__global__ void tl_finalize(const unsigned* __restrict__ P, const unsigned* __restrict__ M,
                            const int* __restrict__ cnt, float* __restrict__ out) {
    int c = threadIdx.x;            // 32 threads (one wave)
    float v = 0.0f;
    if (c < NCLS) {
        float p = __uint_as_float(P[c]);
        float m = __uint_as_float(M[c]);
        v = (float)cnt[c] * fmaxf(p - m + MARGIN, 0.0f);
    }
    #pragma unroll
    for (int off = 16; off > 0; off >>= 1) v += __shfl_down(v, off, 32);
    if (c == 0) out[0] = v;
}

extern "C" void kernel_launch(void* const* d_in, const int* in_sizes, int n_in,
                              void* d_out, int out_size, void* d_ws, size_t ws_size,
                              hipStream_t stream) {
    const float* emb    = (const float*)d_in[0];   // [8192, 64] f32
    const int*   labels = (const int*)d_in[1];     // [8192] int
    const float* labemb = (const float*)d_in[2];   // [20, 64] f32
    float*       out    = (float*)d_out;

    unsigned* P   = (unsigned*)d_ws;               // 32
    unsigned* M   = P + NPAD;                      // 32
    int*      cnt = (int*)(M + NPAD);              // 32

    tl_init_ws<<<1, 32, 0, stream>>>(P, M, cnt);
    tl_count<<<NROWS / 256, 256, 0, stream>>>(labels, cnt);
    tl_dist_reduce<<<NROWS / 128, 256, 0, stream>>>(emb, labels, labemb, P, M);
    tl_finalize<<<1, 32, 0, stream>>>(P, M, cnt, out);
}